// EntropicLayer_63574105916111
// MI455X (gfx1250) — compile-verified
//
#include <hip/hip_runtime.h>
#include <math.h>

#define N_NODES 100000
#define N_EDGES 1000000
#define D_IN    128
#define D_OUT   64

typedef float v2f __attribute__((ext_vector_type(2)));
typedef float v8f __attribute__((ext_vector_type(8)));

// ---------------------------------------------------------------------------
// helpers
// ---------------------------------------------------------------------------
__device__ __forceinline__ float waveReduceSum(float v) {
#pragma unroll
  for (int off = 16; off > 0; off >>= 1) v += __shfl_xor(v, off, 32);
  return v;
}

__device__ __forceinline__ float waveReduceMax(float v) {
#pragma unroll
  for (int off = 16; off > 0; off >>= 1) v = fmaxf(v, __shfl_xor(v, off, 32));
  return v;
}

// two-sided int-trick float atomic max (init target to -INF)
__device__ __forceinline__ void atomicMaxFloat(float* addr, float value) {
  if (value >= 0.0f)
    atomicMax((int*)addr, __float_as_int(value));
  else
    atomicMin((unsigned int*)addr, __float_as_uint(value));
}

// ---------------------------------------------------------------------------
// 1) init: deg = 1 (self loop), e = 0, reduction scalars
// ---------------------------------------------------------------------------
__global__ void k_init(float* deg, float* e, float* red) {
  int i = blockIdx.x * blockDim.x + threadIdx.x;
  if (i < N_NODES) { deg[i] = 1.0f; e[i] = 0.0f; }
  if (i == 0) { red[0] = -INFINITY; red[1] = 0.0f; red[2] = 0.0f; }
}

// ---------------------------------------------------------------------------
// 2) degree histogram over dst
// ---------------------------------------------------------------------------
__global__ void k_degree(const int* __restrict__ ei, float* deg) {
  int i = blockIdx.x * blockDim.x + threadIdx.x;
  if (i < N_EDGES) atomicAdd(&deg[ei[N_EDGES + i]], 1.0f);
}

// ---------------------------------------------------------------------------
// 3) h0 = x @ W via V_WMMA_F32_16X16X4_F32 (wave32, one 16x16 tile per wave)
// ---------------------------------------------------------------------------
__global__ __launch_bounds__(256) void k_gemm_wmma(const float* __restrict__ x,
                                                   const float* __restrict__ W,
                                                   float* __restrict__ h0) {
  __shared__ float sW[D_IN * D_OUT];  // 32 KB of the 320 KB WGP LDS
  for (int i = threadIdx.x; i < D_IN * D_OUT; i += 256) sW[i] = W[i];
  __syncthreads();

  const int lane  = threadIdx.x & 31;
  const int wave  = threadIdx.x >> 5;
  const int tile  = blockIdx.x * 8 + wave;      // 25000 tiles exactly
  const int mtile = tile >> 2;                  // 6250 M tiles
  const int ntile = tile & 3;                   // 4 N tiles

  // A 16x4 fragment: lanes 0-15 -> K={k0,k0+1}, lanes 16-31 -> K={k0+2,k0+3}
  const int kb = (lane >> 4) * 2;
  const int n  = ntile * 16 + (lane & 15);
  const float* __restrict__ arow = x + (size_t)(mtile * 16 + (lane & 15)) * D_IN;

  v8f acc = {0.f, 0.f, 0.f, 0.f, 0.f, 0.f, 0.f, 0.f};
#pragma unroll 4
  for (int k0 = 0; k0 < D_IN; k0 += 4) {
    v2f a, b;
    a.x = arow[k0 + kb];
    a.y = arow[k0 + kb + 1];
    b.x = sW[(k0 + kb)     * D_OUT + n];
    b.y = sW[(k0 + kb + 1) * D_OUT + n];
    acc = __builtin_amdgcn_wmma_f32_16x16x4_f32(
        /*neg_a=*/false, a, /*neg_b=*/false, b,
        /*c_mod=*/(short)0, acc, /*reuse_a=*/false, /*reuse_b=*/false);
  }

  // C/D layout: VGPR i -> M = i (lanes 0-15) / 8+i (lanes 16-31), N = lane&15
  const int crow = mtile * 16 + (lane >> 4) * 8;
  float* __restrict__ outp = h0 + (size_t)crow * D_OUT + n;
#pragma unroll
  for (int i = 0; i < 8; ++i) outp[(size_t)i * D_OUT] = acc[i];
}

// ---------------------------------------------------------------------------
// 4) h = h0 / deg + b   (self-loop term; agg added on top by k_agg)
// ---------------------------------------------------------------------------
__global__ void k_selfloop(const float* __restrict__ h0, const float* __restrict__ deg,
                           const float* __restrict__ b, float* __restrict__ h) {
  int i = blockIdx.x * blockDim.x + threadIdx.x;
  if (i < N_NODES * D_OUT) {
    int node = i >> 6, j = i & 63;
    h[i] = h0[i] / deg[node] + b[j];
  }
}

// ---------------------------------------------------------------------------
// 5) h[dst] += rsqrt(deg_s) * rsqrt(deg_d) * h0[src]   (edge x col scatter)
// ---------------------------------------------------------------------------
__global__ void k_agg(const int* __restrict__ ei, const float* __restrict__ h0,
                      const float* __restrict__ deg, float* __restrict__ h) {
  int i = blockIdx.x * blockDim.x + threadIdx.x;
  if (i >= N_EDGES * D_OUT) return;
  int e = i >> 6, j = i & 63;
  int s = ei[e], d = ei[N_EDGES + e];
  float norm = rsqrtf(deg[s]) * rsqrtf(deg[d]);
  atomicAdd(&h[(size_t)d * D_OUT + j], norm * h0[(size_t)s * D_OUT + j]);
}

// ---------------------------------------------------------------------------
// 6) e[dst] += ||h[dst] - h[src]||^2   (one wave per edge)
// ---------------------------------------------------------------------------
__global__ __launch_bounds__(256) void k_energy(const int* __restrict__ ei,
                                                const float* __restrict__ h,
                                                float* __restrict__ e) {
  int lane = threadIdx.x & 31;
  int edge = blockIdx.x * 8 + (threadIdx.x >> 5);
  if (edge >= N_EDGES) return;
  int s = ei[edge], d = ei[N_EDGES + edge];
  const float* hs = h + (size_t)s * D_OUT;
  const float* hd = h + (size_t)d * D_OUT;
  float d0 = hd[lane]      - hs[lane];
  float d1 = hd[lane + 32] - hs[lane + 32];
  float v  = waveReduceSum(d0 * d0 + d1 * d1);
  if (lane == 0) atomicAdd(&e[d], v);
}

// ---------------------------------------------------------------------------
// 7) global max of s = -e/T
// ---------------------------------------------------------------------------
__global__ void k_max(const float* __restrict__ e, const float* __restrict__ Tp,
                      float* red) {
  int i = blockIdx.x * blockDim.x + threadIdx.x;
  float T = *Tp;
  float s = (i < N_NODES) ? (-e[i] / T) : -INFINITY;
  s = waveReduceMax(s);
  if ((threadIdx.x & 31) == 0) atomicMaxFloat(&red[0], s);
}

// ---------------------------------------------------------------------------
// 8) S1 = sum exp(s-M), S2 = sum s*exp(s-M)
// ---------------------------------------------------------------------------
__global__ void k_sumexp(const float* __restrict__ e, const float* __restrict__ Tp,
                         float* red) {
  int i = blockIdx.x * blockDim.x + threadIdx.x;
  float T = *Tp, M = red[0];
  float w = 0.0f, sw = 0.0f;
  if (i < N_NODES) {
    float s = -e[i] / T;
    w  = __expf(s - M);
    sw = s * w;
  }
  w  = waveReduceSum(w);
  sw = waveReduceSum(sw);
  if ((threadIdx.x & 31) == 0) { atomicAdd(&red[1], w); atomicAdd(&red[2], sw); }
}

// ---------------------------------------------------------------------------
// 9) cT[n] = p_n * (logp_n + H)   (= temperature-scaled dH/de * T)
// ---------------------------------------------------------------------------
__global__ void k_coef(const float* __restrict__ e, const float* __restrict__ Tp,
                       const float* __restrict__ red, float* __restrict__ cT) {
  int i = blockIdx.x * blockDim.x + threadIdx.x;
  if (i >= N_NODES) return;
  float T = *Tp;
  float M = red[0], S1 = red[1], S2 = red[2];
  float logZ = M + __logf(S1);
  float H    = logZ - S2 / S1;          // H = logZ - sum(p*s)
  float s    = -e[i] / T;
  float logp = s - logZ;
  cT[i] = __expf(logp) * (logp + H);
}

// ---------------------------------------------------------------------------
// 10) out += weight * T*grad: per edge +-2*cT[d]*(h_d - h_s) (out pre-set = h)
// ---------------------------------------------------------------------------
__global__ void k_grad(const int* __restrict__ ei, const float* __restrict__ h,
                       const float* __restrict__ cT, const float* __restrict__ Wp,
                       float* __restrict__ out) {
  int i = blockIdx.x * blockDim.x + threadIdx.x;
  if (i >= N_EDGES * D_OUT) return;
  int e = i >> 6, j = i & 63;
  int s = ei[e], d = ei[N_EDGES + e];
  float coef  = 2.0f * (*Wp) * cT[d];
  float delta = h[(size_t)d * D_OUT + j] - h[(size_t)s * D_OUT + j];
  float contrib = coef * delta;
  atomicAdd(&out[(size_t)d * D_OUT + j],  contrib);
  atomicAdd(&out[(size_t)s * D_OUT + j], -contrib);
}

// ---------------------------------------------------------------------------
extern "C" void kernel_launch(void* const* d_in, const int* in_sizes, int n_in,
                              void* d_out, int out_size, void* d_ws, size_t ws_size,
                              hipStream_t stream) {
  const float* x  = (const float*)d_in[0];   // [N, 128]
  const int*   ei = (const int*)  d_in[1];   // [2, E] flat: src then dst
  const float* wp = (const float*)d_in[2];   // scalar weight
  const float* tp = (const float*)d_in[3];   // scalar temperature
  const float* W  = (const float*)d_in[4];   // [128, 64]
  const float* b  = (const float*)d_in[5];   // [64]
  float* out = (float*)d_out;                // [N, 64]

  // workspace layout (floats)
  float* h0  = (float*)d_ws;                 // N*64
  float* h   = h0  + (size_t)N_NODES * D_OUT;// N*64
  float* deg = h   + (size_t)N_NODES * D_OUT;// N
  float* e   = deg + N_NODES;                // N
  float* cT  = e   + N_NODES;                // N
  float* red = cT  + N_NODES;                // 3 scalars: M, S1, S2

  const int TB = 256;
  const int nodeBlocks = (N_NODES + TB - 1) / TB;           // 391
  const int edgeBlocks = (N_EDGES + TB - 1) / TB;           // 3907
  const int ecolBlocks = (N_EDGES * D_OUT) / TB;            // 250000 (exact)
  const int ncolBlocks = (N_NODES * D_OUT) / TB;            // 25000  (exact)
  const int gemmBlocks = (N_NODES / 16) * (D_OUT / 16) / 8; // 3125   (exact)
  const int waveEdgeBlocks = N_EDGES / 8;                   // 125000 (exact)

  k_init     <<<nodeBlocks,     TB, 0, stream>>>(deg, e, red);
  k_degree   <<<edgeBlocks,     TB, 0, stream>>>(ei, deg);
  k_gemm_wmma<<<gemmBlocks,     TB, 0, stream>>>(x, W, h0);
  k_selfloop <<<ncolBlocks,     TB, 0, stream>>>(h0, deg, b, h);
  k_agg      <<<ecolBlocks,     TB, 0, stream>>>(ei, h0, deg, h);
  k_energy   <<<waveEdgeBlocks, TB, 0, stream>>>(ei, h, e);
  k_max      <<<nodeBlocks,     TB, 0, stream>>>(e, tp, red);
  k_sumexp   <<<nodeBlocks,     TB, 0, stream>>>(e, tp, red);
  k_coef     <<<nodeBlocks,     TB, 0, stream>>>(e, tp, red, cT);
  hipMemcpyAsync(out, h, (size_t)N_NODES * D_OUT * sizeof(float),
                 hipMemcpyDeviceToDevice, stream);
  k_grad     <<<ecolBlocks,     TB, 0, stream>>>(ei, h, cT, wp, out);
}